// ConvolutionalSpatialTransformer_12378095747614
// MI455X (gfx1250) — compile-verified
//
#include <hip/hip_runtime.h>
#include <hip/hip_bf16.h>

typedef float v2f __attribute__((ext_vector_type(2)));
typedef float v8f __attribute__((ext_vector_type(8)));

#define NB 4
#define HB 480
#define WB 640

#define PIX_PER_BLOCK 128                 // 8 waves x 16 pixels, one row chunk
#define HALO_W (PIX_PER_BLOCK + 4)        // 132 columns (5x5 conv, pad 2)
#define CHUNKS_PER_ROW (WB / PIX_PER_BLOCK)  // 5

// LDS offset of input element for GEMM-K index k (compile-time after unroll):
// k = ic*25 + ky*5 + kx  ->  inLds[ic][ky][col + kx], row pitch HALO_W, ch pitch 5*HALO_W
__device__ constexpr int ldsOffOfK(int k) {
    const int ic = k / 25;
    const int r  = k - ic * 25;
    const int ky = r / 5;
    const int kx = r - ky * 5;
    return ic * (5 * HALO_W) + ky * HALO_W + kx;
}

// ---------------------------------------------------------------------------
// 5x5 conv, pad 2, CIN -> 4 channels, implicit GEMM on V_WMMA_F32_16X16X4_F32.
// Per block: stage [CIN][5][132] input halo + [16][KC] zero-padded weights in
// LDS, then each wave32 runs a fully-unrolled K loop of WMMAs over its 16
// pixels. Fragment layouts per ISA 7.12.2:
//   A 16x4: lane m = M (rows 4..15 zero), VGPR t = K{kb+khalf+t}, khalf=2*(lane>=16)
//   B 4x16: lane n = pixel (lane&15), VGPR t = K{kb+khalf+t}
//   C/D:    acc[r] = row M=r for lanes 0..15 -> out-channels 0..3 in acc[0..3]
// ---------------------------------------------------------------------------
template <int CIN>
__global__ __launch_bounds__(256) void conv5x5_wmma_kernel(
    const float* __restrict__ in, const float* __restrict__ w,
    const float* __restrict__ bias, float* __restrict__ out)
{
    constexpr int K  = CIN * 25;
    constexpr int KC = (K + 3) & ~3;

    __shared__ float wLds[16 * KC];
    __shared__ float inLds[CIN * 5 * HALO_W];

    // ---- block coordinates: one (n, y, 128-pixel chunk) per block ----
    const int b    = blockIdx.x;
    const int ch   = b % CHUNKS_PER_ROW;
    const int y    = (b / CHUNKS_PER_ROW) % HB;
    const int n    = b / (CHUNKS_PER_ROW * HB);
    const int xblk = ch * PIX_PER_BLOCK;
    const long plane = (long)HB * WB;

    // ---- stage weights: wLds[m][k] = w[m,k] for m<4 && k<K else 0 ----
    for (int i = threadIdx.x; i < 16 * KC; i += blockDim.x) {
        const int mm = i / KC;
        const int k  = i - mm * KC;
        wLds[i] = (mm < 4 && k < K) ? w[mm * K + k] : 0.f;
    }
    // ---- stage input halo with zero padding ----
    for (int i = threadIdx.x; i < CIN * 5 * HALO_W; i += blockDim.x) {
        const int c  = i % HALO_W;
        const int rr = (i / HALO_W) % 5;
        const int ic = i / (5 * HALO_W);
        const int gy = y + rr - 2;
        const int gx = xblk + c - 2;
        float v = 0.f;
        if (gy >= 0 && gy < HB && gx >= 0 && gx < WB)
            v = in[((long)n * CIN + ic) * plane + (long)gy * WB + gx];
        inLds[i] = v;
    }
    __syncthreads();

    // ---- per-wave GEMM ----
    const int lane  = threadIdx.x & 31;
    const int wave  = threadIdx.x >> 5;
    const int pix   = lane & 15;
    const bool hiHalf = (lane >= 16);
    const int khalf = hiHalf ? 2 : 0;
    const int m     = lane & 15;
    const int aBase = m * KC + khalf;           // + (kb + t) immediate
    const int bBase = wave * 16 + pix;          // + ldsOffOfK(k) immediate (selected)

    v8f acc = {};
#pragma unroll
    for (int kb = 0; kb < KC; kb += 4) {
        v2f afrag, bfrag;
#pragma unroll
        for (int t = 0; t < 2; ++t) {
            // lanes 0..15 want k = kb+t, lanes 16..31 want k = kb+2+t
            constexpr int dummy = 0; (void)dummy;
            const int kLo = (kb + t     < K) ? (kb + t)     : (K - 1);
            const int kHi = (kb + 2 + t < K) ? (kb + 2 + t) : (K - 1);
            const int oLo = ldsOffOfK(kLo);     // compile-time constants
            const int oHi = ldsOffOfK(kHi);
            const int o   = hiHalf ? oHi : oLo; // one v_cndmask
            bfrag[t] = inLds[bBase + o];
            afrag[t] = wLds[aBase + kb + t];    // contiguous pair -> b64
        }
        // EXEC is all-1s here (no divergent control flow in this phase).
        acc = __builtin_amdgcn_wmma_f32_16x16x4_f32(
            false, afrag, false, bfrag, (short)0, acc, false, false);
    }

    // lanes 0..15 hold rows M=0..7 in acc[0..7]; channels are M=0..3
    if (lane < 16) {
        const int px = xblk + wave * 16 + pix;
#pragma unroll
        for (int oc = 0; oc < 4; ++oc)
            out[((long)n * 4 + oc) * plane + (long)y * WB + px] = acc[oc] + bias[oc];
    }
}

// ---------------------------------------------------------------------------
// Fused: grid construction + bilinear grid_sample (zero pad, reference's
// y/x channel-swap quirk preserved) + 3x3 stride-3 conv. One thread = one
// output pixel; reads theta[n,:,y,x] once, gathers 9 samples x 3 ch x 4 taps.
// ---------------------------------------------------------------------------
__global__ __launch_bounds__(256) void warp_resample_conv_kernel(
    const float* __restrict__ x, const float* __restrict__ theta,
    const float* __restrict__ wr, const float* __restrict__ br,
    float* __restrict__ out)
{
    __shared__ float sw[81];
    __shared__ float sb[3];
    if (threadIdx.x < 81) sw[threadIdx.x] = wr[threadIdx.x];
    if (threadIdx.x < 3)  sb[threadIdx.x] = br[threadIdx.x];
    __syncthreads();

    const long total = (long)NB * HB * WB;
    const long idx   = (long)blockIdx.x * blockDim.x + threadIdx.x;
    if (idx >= total) return;

    const int px = (int)(idx % WB);
    int rem      = (int)(idx / WB);
    const int py = rem % HB;
    const int n  = rem / HB;

    const long plane  = (long)HB * WB;
    const long thbase = (long)n * 4 * plane + (long)py * WB + px;
    const float t0 = theta[thbase];
    const float t1 = theta[thbase + plane];
    const float t2 = theta[thbase + 2 * plane];
    const float t3 = theta[thbase + 3 * plane];

    // G0 channel 0 = yy[py], channel 1 = xx[px]  (linspace(-1,1,·))
    const float g0y = -1.f + 2.f * (float)py / (float)(HB - 1);
    const float g0x = -1.f + 2.f * (float)px / (float)(WB - 1);
    // local s=3 pattern: ly = linspace(-3/H, 3/H, 3), lx = linspace(-3/W, 3/W, 3)
    const float ly[3] = { -3.f / (float)HB, 0.f, 3.f / (float)HB };
    const float lx[3] = { -3.f / (float)WB, 0.f, 3.f / (float)WB };

    float acc0 = sb[0], acc1 = sb[1], acc2 = sb[2];

#pragma unroll
    for (int ky = 0; ky < 3; ++ky) {
#pragma unroll
        for (int kx = 0; kx < 3; ++kx) {
            // Reference quirk: grid channel 0 (built from yy + theta0/1) is the
            // X coordinate in grid_sample; channel 1 (xx + theta2/3) is Y.
            const float gx = g0y + t0 * ly[ky] + t1 * lx[kx];
            const float gy = g0x + t2 * ly[ky] + t3 * lx[kx];
            const float fx = ((gx + 1.f) * (float)WB - 1.f) * 0.5f;
            const float fy = ((gy + 1.f) * (float)HB - 1.f) * 0.5f;
            const float fx0 = floorf(fx), fy0 = floorf(fy);
            const float wx1 = fx - fx0,  wy1 = fy - fy0;
            const float wx0 = 1.f - wx1, wy0 = 1.f - wy1;
            const int ix0 = (int)fx0, iy0 = (int)fy0;
            const bool vx0 = (ix0 >= 0) && (ix0 < WB);
            const bool vx1 = (ix0 + 1 >= 0) && (ix0 + 1 < WB);
            const bool vy0 = (iy0 >= 0) && (iy0 < HB);
            const bool vy1 = (iy0 + 1 >= 0) && (iy0 + 1 < HB);
            const int cx0 = min(max(ix0, 0), WB - 1);
            const int cx1 = min(max(ix0 + 1, 0), WB - 1);
            const int cy0 = min(max(iy0, 0), HB - 1);
            const int cy1 = min(max(iy0 + 1, 0), HB - 1);
            const float w00 = wx0 * wy0, w01 = wx1 * wy0;
            const float w10 = wx0 * wy1, w11 = wx1 * wy1;
#pragma unroll
            for (int ic = 0; ic < 3; ++ic) {
                const float* img = x + ((long)n * 3 + ic) * plane;
                const float v00 = (vx0 && vy0) ? img[(long)cy0 * WB + cx0] : 0.f;
                const float v01 = (vx1 && vy0) ? img[(long)cy0 * WB + cx1] : 0.f;
                const float v10 = (vx0 && vy1) ? img[(long)cy1 * WB + cx0] : 0.f;
                const float v11 = (vx1 && vy1) ? img[(long)cy1 * WB + cx1] : 0.f;
                const float val = v00 * w00 + v01 * w01 + v10 * w10 + v11 * w11;
                acc0 += sw[((0 * 3 + ic) * 3 + ky) * 3 + kx] * val;
                acc1 += sw[((1 * 3 + ic) * 3 + ky) * 3 + kx] * val;
                acc2 += sw[((2 * 3 + ic) * 3 + ky) * 3 + kx] * val;
            }
        }
    }

    const long obase = (long)n * 3 * plane + (long)py * WB + px;
    out[obase]             = acc0;
    out[obase + plane]     = acc1;
    out[obase + 2 * plane] = acc2;
}

extern "C" void kernel_launch(void* const* d_in, const int* in_sizes, int n_in,
                              void* d_out, int out_size, void* d_ws, size_t ws_size,
                              hipStream_t stream)
{
    (void)in_sizes; (void)n_in; (void)out_size; (void)ws_size;
    const float* x  = (const float*)d_in[0];
    const float* w1 = (const float*)d_in[1];
    const float* b1 = (const float*)d_in[2];
    const float* w2 = (const float*)d_in[3];
    const float* b2 = (const float*)d_in[4];
    const float* w3 = (const float*)d_in[5];
    const float* b3 = (const float*)d_in[6];
    const float* w4 = (const float*)d_in[7];
    const float* b4 = (const float*)d_in[8];
    const float* wr = (const float*)d_in[9];
    const float* br = (const float*)d_in[10];
    // d_in[11] = grid_size, fixed at 3 by the reference setup.

    // Ping-pong activation buffers in scratch: 2 x [N,4,H,W] f32 = 39.3 MB.
    const size_t actElems = (size_t)NB * 4 * HB * WB;
    float* bufA = (float*)d_ws;
    float* bufB = bufA + actElems;

    const int convBlocks = NB * HB * CHUNKS_PER_ROW;   // 9600 blocks x 256 thr

    conv5x5_wmma_kernel<3><<<convBlocks, 256, 0, stream>>>(x,    w1, b1, bufA);
    conv5x5_wmma_kernel<4><<<convBlocks, 256, 0, stream>>>(bufA, w2, b2, bufB);
    conv5x5_wmma_kernel<4><<<convBlocks, 256, 0, stream>>>(bufB, w3, b3, bufA);
    conv5x5_wmma_kernel<4><<<convBlocks, 256, 0, stream>>>(bufA, w4, b4, bufB);

    const long totalPix = (long)NB * HB * WB;
    const int  wBlocks  = (int)((totalPix + 255) / 256);
    warp_resample_conv_kernel<<<wBlocks, 256, 0, stream>>>(x, bufB, wr, br, (float*)d_out);
}